// SABlock_78675210928562
// MI455X (gfx1250) — compile-verified
//
#include <hip/hip_runtime.h>
#include <math.h>

typedef __attribute__((ext_vector_type(2))) float v2f;
typedef __attribute__((ext_vector_type(8))) float v8f;

#define NN 4
#define LL 256
#define DD 128
#define PP 64
#define CCH 4
#define NPAIR (NN*LL*LL)            /* 262144 */
#define KMSG  (2*DD + CCH*CCH + PP) /* 336 */
#define KEDGE (2*DD + PP)           /* 320 */

static __device__ __forceinline__ v8f wmma4(v2f a, v2f b, v8f c) {
  // D = A(16x4,f32) * B(4x16,f32) + C(16x16,f32)
  return __builtin_amdgcn_wmma_f32_16x16x4_f32(false, a, false, b, (short)0, c, false, false);
}

static __device__ __forceinline__ float silu(float x) { return x / (1.f + __expf(-x)); }

// Async memory -> LDS copy (16B per active lane). vdst VGPR holds the wave-relative
// LDS byte address (low 32 bits of a generic LDS pointer), vaddr the 64-bit global addr.
static __device__ __forceinline__ void async_lds_b128(unsigned lds_byte, const float* gptr) {
  asm volatile("global_load_async_to_lds_b128 %0, %1, off"
               :: "v"(lds_byte), "v"(gptr) : "memory");
}
static __device__ __forceinline__ void wait_async0() {
#if __has_builtin(__builtin_amdgcn_s_wait_asynccnt)
  __builtin_amdgcn_s_wait_asynccnt(0);
#else
  asm volatile("s_wait_asynccnt 0x0" ::: "memory");
#endif
}
static __device__ __forceinline__ unsigned lds_off(const void* p) {
  return (unsigned)(size_t)p;  // flat LDS address: addr[31:0] == wave-relative LDS offset
}

// ---------------- K0: weight transpose [K][C] -> [C][K] ----------------
__global__ void k_transpose(const float* __restrict__ src, float* __restrict__ dst,
                            int K, int C) {
  int idx = blockIdx.x * 256 + threadIdx.x;
  if (idx < K * C) {
    int k = idx / C, c = idx - k * C;
    dst[(size_t)c * K + k] = src[idx];
  }
}

// ---------------- K1: cd, raw radial, per-block partial sums of radial^2 ----------------
__global__ void k_cd_radial(const float* __restrict__ coord,
                            float* __restrict__ cd, float* __restrict__ radial,
                            float* __restrict__ partials) {
  int pair = blockIdx.x * 256 + threadIdx.x;
  int j = pair & 255, t2 = pair >> 8, i = t2 & 255, n = t2 >> 8;
  const float* ci = coord + (size_t)(n*LL + i) * 12;
  const float* cj = coord + (size_t)(n*LL + j) * 12;
  float d[12];
#pragma unroll
  for (int q = 0; q < 12; ++q) d[q] = ci[q] - cj[q];
  float* cdp = cd + (size_t)pair * 12;
#pragma unroll
  for (int q = 0; q < 12; ++q) cdp[q] = d[q];
  float sq[16];
  float* rp = radial + (size_t)pair * 16;
#pragma unroll
  for (int a = 0; a < 4; ++a)
#pragma unroll
    for (int b = 0; b < 4; ++b) {
      float s = d[a*3]*d[b*3] + d[a*3+1]*d[b*3+1] + d[a*3+2]*d[b*3+2];
      rp[a*4 + b] = s;
      sq[a*4 + b] = s * s;
    }
  __shared__ float red[256];
  for (int q = 0; q < 16; ++q) {
    red[threadIdx.x] = sq[q];
    __syncthreads();
    for (int s = 128; s > 0; s >>= 1) {
      if (threadIdx.x < s) red[threadIdx.x] += red[threadIdx.x + s];
      __syncthreads();
    }
    if (threadIdx.x == 0) partials[(size_t)blockIdx.x * 16 + q] = red[0];
    __syncthreads();
  }
}

// ---------------- K2: final norm reduction -> reciprocal norms ----------------
__global__ void k_norm(const float* __restrict__ partials, float* __restrict__ rnorm) {
  int t = threadIdx.x;
  if (t < 16) {
    float s = 0.f;
    for (int b = 0; b < NPAIR/256; ++b) s += partials[(size_t)b*16 + t];
    float nv = sqrtf(s);
    rnorm[t] = 1.f / fmaxf(nv, 1e-12f);
  }
}

// ---------------- K3: message MLP (336 -> 128 SiLU -> 128) via f32 WMMA ----------------
// A-tile h_i / h_j / z segments staged by async DMA (b128); radial segment scaled in VALU.
__global__ void k_msg(const float* __restrict__ h, const float* __restrict__ z,
                      const float* __restrict__ radial, const float* __restrict__ rnorm,
                      const float* __restrict__ w1t, const float* __restrict__ b1,
                      const float* __restrict__ w2t, const float* __restrict__ b2,
                      float* __restrict__ m) {
  __shared__ __align__(16) float A[16 * KMSG];
  __shared__ __align__(16) float H[16 * DD];
  int rowBase = blockIdx.x * 16;
  int tid = threadIdx.x;
  unsigned Abase = lds_off(A);
  // 80 x b128 transfers per row (h_i:32, h_j:32, z:16), 16 rows => 1280 = 5*256
  for (int t = tid; t < 16 * 80; t += 256) {
    int row = t / 80, s = t - row * 80;
    int r = rowBase + row;
    int j = r & 255, t2 = r >> 8, i = t2 & 255, n = t2 >> 8;
    unsigned doff;
    const float* g;
    if (s < 32)      { doff = Abase + (unsigned)(row*KMSG + s*4) * 4u;
                       g = h + (size_t)(n*LL + i)*DD + s*4; }
    else if (s < 64) { doff = Abase + (unsigned)(row*KMSG + 128 + (s-32)*4) * 4u;
                       g = h + (size_t)(n*LL + j)*DD + (s-32)*4; }
    else             { doff = Abase + (unsigned)(row*KMSG + 272 + (s-64)*4) * 4u;
                       g = z + (size_t)r*PP + (s-64)*4; }
    async_lds_b128(doff, g);
  }
  {  // radial segment: 16 rows x 16 entries, scaled by rnorm
    int row = tid >> 4, q = tid & 15;
    int r = rowBase + row;
    A[row*KMSG + 2*DD + q] = radial[(size_t)r*16 + q] * rnorm[q];
  }
  wait_async0();
  __syncthreads();

  int wave = tid >> 5, lane = tid & 31, l15 = lane & 15, kOff = (lane >> 4) * 2;
  int col = wave * 16 + l15;
  v8f acc = {};
  const float* wcol1 = w1t + (size_t)col * KMSG;
  for (int k0 = 0; k0 < KMSG; k0 += 4) {
    v2f a = *(const v2f*)&A[l15*KMSG + k0 + kOff];
    v2f b = *(const v2f*)&wcol1[k0 + kOff];
    acc = wmma4(a, b, acc);
  }
  float bias1 = b1[col];
#pragma unroll
  for (int g = 0; g < 8; ++g) {
    int rrow = g + ((lane >> 4) << 3);
    H[rrow*DD + col] = silu(acc[g] + bias1);
  }
  __syncthreads();
  v8f acc2 = {};
  const float* wcol2 = w2t + (size_t)col * DD;
  for (int k0 = 0; k0 < DD; k0 += 4) {
    v2f a = *(const v2f*)&H[l15*DD + k0 + kOff];
    v2f b = *(const v2f*)&wcol2[k0 + kOff];
    acc2 = wmma4(a, b, acc2);
  }
  float bias2 = b2[col];
#pragma unroll
  for (int g = 0; g < 8; ++g) {
    int rrow = g + ((lane >> 4) << 3);
    int r = rowBase + rrow;
    m[(size_t)r*DD + col] = acc2[g] + bias2;
  }
}

// ---------------- K4: masked aggregation over j ----------------
__global__ void k_agg(const float* __restrict__ m, const float* __restrict__ intra,
                      const float* __restrict__ inter,
                      float* __restrict__ aggi, float* __restrict__ agge) {
  int row = blockIdx.x;  // n*L + i
  int d = threadIdx.x;   // 128
  const float* mp = m + (size_t)row * LL * DD;
  const float* mi = intra + (size_t)row * LL;
  const float* me = inter + (size_t)row * LL;
  float si = 0.f, se = 0.f;
  for (int j = 0; j < LL; ++j) {
    float v = mp[(size_t)j*DD + d];
    si += v * mi[j];
    se += v * me[j];
  }
  aggi[(size_t)row*DD + d] = si;
  agge[(size_t)row*DD + d] = se;
}

// ---------------- K5: node update + LayerNorm ----------------
__global__ void k_node(const float* __restrict__ h,
                       const float* __restrict__ aggi, const float* __restrict__ agge,
                       const float* __restrict__ rel_w0, const float* __restrict__ rel_w1,
                       const float* __restrict__ nw1, const float* __restrict__ nb1,
                       const float* __restrict__ nw2, const float* __restrict__ nb2,
                       const float* __restrict__ ln_w, const float* __restrict__ ln_b,
                       float* __restrict__ h_new) {
  int row = blockIdx.x;  // n*L + i
  int d = threadIdx.x;   // 128
  __shared__ float nin[3*DD];
  __shared__ float ai[DD];
  __shared__ float ae[DD];
  __shared__ float hid[DD];
  __shared__ float red[DD];
  float hv = h[(size_t)row*DD + d];
  nin[d] = hv;
  ai[d] = aggi[(size_t)row*DD + d];
  ae[d] = agge[(size_t)row*DD + d];
  __syncthreads();
  float s0 = 0.f, s1 = 0.f;
  for (int k = 0; k < DD; ++k) {
    s0 += ai[k] * rel_w0[(size_t)k*DD + d];
    s1 += ae[k] * rel_w1[(size_t)k*DD + d];
  }
  nin[DD + d] = s0;
  nin[2*DD + d] = s1;
  __syncthreads();
  float a1 = nb1[d];
  for (int k = 0; k < 3*DD; ++k) a1 += nin[k] * nw1[(size_t)k*DD + d];
  hid[d] = silu(a1);
  __syncthreads();
  float u = nb2[d];
  for (int k = 0; k < DD; ++k) u += hid[k] * nw2[(size_t)k*DD + d];
  float x = hv + u;
  red[d] = x;
  __syncthreads();
  for (int s = 64; s > 0; s >>= 1) {
    if (d < s) red[d] += red[d + s];
    __syncthreads();
  }
  float mu = red[0] * (1.f / 128.f);
  __syncthreads();
  float dx = x - mu;
  red[d] = dx * dx;
  __syncthreads();
  for (int s = 64; s > 0; s >>= 1) {
    if (d < s) red[d] += red[d + s];
    __syncthreads();
  }
  float var = red[0] * (1.f / 128.f);
  h_new[(size_t)row*DD + d] = dx * rsqrtf(var + 1e-5f) * ln_w[d] + ln_b[d];
}

// ---------------- K6: edge MLP (320 -> 128 SiLU -> 64) via f32 WMMA ----------------
__global__ void k_edge(const float* __restrict__ hn, const float* __restrict__ z,
                       const float* __restrict__ w1t, const float* __restrict__ b1,
                       const float* __restrict__ w2t, const float* __restrict__ b2,
                       float* __restrict__ z_new) {
  __shared__ __align__(16) float A[16 * KEDGE];
  __shared__ __align__(16) float H[16 * DD];
  int rowBase = blockIdx.x * 16;
  int tid = threadIdx.x;
  unsigned Abase = lds_off(A);
  // 80 x b128 per row (h_i:32, h_j:32, z:16) -> 1280 transfers, fully async
  for (int t = tid; t < 16 * 80; t += 256) {
    int row = t / 80, s = t - row * 80;
    int r = rowBase + row;
    int j = r & 255, t2 = r >> 8, i = t2 & 255, n = t2 >> 8;
    unsigned doff;
    const float* g;
    if (s < 32)      { doff = Abase + (unsigned)(row*KEDGE + s*4) * 4u;
                       g = hn + (size_t)(n*LL + i)*DD + s*4; }
    else if (s < 64) { doff = Abase + (unsigned)(row*KEDGE + 128 + (s-32)*4) * 4u;
                       g = hn + (size_t)(n*LL + j)*DD + (s-32)*4; }
    else             { doff = Abase + (unsigned)(row*KEDGE + 256 + (s-64)*4) * 4u;
                       g = z + (size_t)r*PP + (s-64)*4; }
    async_lds_b128(doff, g);
  }
  wait_async0();
  __syncthreads();

  int wave = tid >> 5, lane = tid & 31, l15 = lane & 15, kOff = (lane >> 4) * 2;
  {
    int col = wave * 16 + l15;
    v8f acc = {};
    const float* wcol1 = w1t + (size_t)col * KEDGE;
    for (int k0 = 0; k0 < KEDGE; k0 += 4) {
      v2f a = *(const v2f*)&A[l15*KEDGE + k0 + kOff];
      v2f b = *(const v2f*)&wcol1[k0 + kOff];
      acc = wmma4(a, b, acc);
    }
    float bias1 = b1[col];
#pragma unroll
    for (int g = 0; g < 8; ++g) {
      int rrow = g + ((lane >> 4) << 3);
      H[rrow*DD + col] = silu(acc[g] + bias1);
    }
  }
  __syncthreads();
  if (wave < 4) {  // 64 output cols -> 4 wave tiles (uniform per-wave branch)
    int col = wave * 16 + l15;
    v8f acc = {};
    const float* wcol2 = w2t + (size_t)col * DD;
    for (int k0 = 0; k0 < DD; k0 += 4) {
      v2f a = *(const v2f*)&H[l15*DD + k0 + kOff];
      v2f b = *(const v2f*)&wcol2[k0 + kOff];
      acc = wmma4(a, b, acc);
    }
    float bias2 = b2[col];
#pragma unroll
    for (int g = 0; g < 8; ++g) {
      int rrow = g + ((lane >> 4) << 3);
      int r = rowBase + rrow;
      z_new[(size_t)r*PP + col] = acc[g] + bias2;
    }
  }
}

// ---------------- K7: coord MLPs (128->128 SiLU via WMMA, ->4 via VALU), mask-combine ----------------
__global__ void k_coordw(const float* __restrict__ m,
                         const float* __restrict__ c0w1t, const float* __restrict__ c0b1,
                         const float* __restrict__ c0w2,
                         const float* __restrict__ c1w1t, const float* __restrict__ c1b1,
                         const float* __restrict__ c1w2,
                         const float* __restrict__ intra, const float* __restrict__ inter,
                         float* __restrict__ wgt) {
  __shared__ __align__(16) float M[16 * DD];
  __shared__ __align__(16) float H0[16 * DD];
  __shared__ __align__(16) float H1[16 * DD];
  __shared__ float Wtmp[16 * 8];
  int rowBase = blockIdx.x * 16;
  int tid = threadIdx.x;
  unsigned Mbase = lds_off(M);
  // 2048 floats = 512 b128 async transfers = 2 per thread
  for (int t = tid; t < 512; t += 256)
    async_lds_b128(Mbase + (unsigned)t * 16u, m + (size_t)rowBase*DD + t*4);
  wait_async0();
  __syncthreads();

  int wave = tid >> 5, lane = tid & 31, l15 = lane & 15, kOff = (lane >> 4) * 2;
  for (int t = wave; t < 16; t += 8) {  // tiles 0..7: coord0 hidden, 8..15: coord1 hidden
    int col = (t & 7) * 16 + l15;
    const float* Wt = (t < 8) ? c0w1t : c1w1t;
    const float* Bv = (t < 8) ? c0b1 : c1b1;
    float* H        = (t < 8) ? H0 : H1;
    v8f acc = {};
    const float* wcol = Wt + (size_t)col * DD;
    for (int k0 = 0; k0 < DD; k0 += 4) {
      v2f a = *(const v2f*)&M[l15*DD + k0 + kOff];
      v2f b = *(const v2f*)&wcol[k0 + kOff];
      acc = wmma4(a, b, acc);
    }
    float bias = Bv[col];
#pragma unroll
    for (int g = 0; g < 8; ++g) {
      int rrow = g + ((lane >> 4) << 3);
      H[rrow*DD + col] = silu(acc[g] + bias);
    }
  }
  __syncthreads();
  if (tid < 128) {  // 16 rows x (4 w0-cols + 4 w1-cols), K=128 dot in VALU
    int rrow = tid >> 3, q = tid & 7;
    const float* W2 = (q < 4) ? c0w2 : c1w2;
    const float* H  = (q < 4) ? H0 : H1;
    int cc = q & 3;
    float s = 0.f;
    for (int k = 0; k < DD; ++k) s += H[rrow*DD + k] * W2[(size_t)k*CCH + cc];
    Wtmp[rrow*8 + q] = s;
  }
  __syncthreads();
  if (tid < 64) {
    int rrow = tid >> 2, cc = tid & 3;
    int r = rowBase + rrow;
    float w0v = Wtmp[rrow*8 + cc], w1v = Wtmp[rrow*8 + 4 + cc];
    wgt[(size_t)r*CCH + cc] = w0v * intra[r] + w1v * inter[r];
  }
}

// ---------------- K8: coord aggregation + denom -> coord_new ----------------
__global__ void k_coordout(const float* __restrict__ coord, const float* __restrict__ cd,
                           const float* __restrict__ wgt,
                           const float* __restrict__ intra, const float* __restrict__ inter,
                           float* __restrict__ coord_new) {
  int row = blockIdx.x;  // n*L + i
  int tid = threadIdx.x; // 256
  __shared__ float red[256];
  __shared__ float cred[256];
  float dm = intra[(size_t)row*LL + tid] + inter[(size_t)row*LL + tid];
  red[tid] = dm;
  __syncthreads();
  for (int s = 128; s > 0; s >>= 1) {
    if (tid < s) red[tid] += red[tid + s];
    __syncthreads();
  }
  float denom = red[0] + (float)LL * 1e-6f;
  int comp = tid >> 4, l16 = tid & 15;  // comp = c*3 + axis, 12 valid
  float part = 0.f;
  if (comp < 12) {
    const float* cdp = cd + (size_t)row * LL * 12;
    const float* wp  = wgt + (size_t)row * LL * 4;
    int c = comp / 3;
    for (int j = l16; j < LL; j += 16)
      part += cdp[(size_t)j*12 + comp] * wp[(size_t)j*4 + c];
  }
  cred[tid] = part;
  __syncthreads();
  for (int s = 8; s > 0; s >>= 1) {
    if (l16 < s) cred[tid] += cred[tid + s];
    __syncthreads();
  }
  if (tid < 12)
    coord_new[(size_t)row*12 + tid] = coord[(size_t)row*12 + tid] + cred[tid*16] / denom;
}

extern "C" void kernel_launch(void* const* d_in, const int* in_sizes, int n_in,
                              void* d_out, int out_size, void* d_ws, size_t ws_size,
                              hipStream_t stream) {
  const float* coord  = (const float*)d_in[2];
  const float* h      = (const float*)d_in[3];
  const float* z      = (const float*)d_in[4];
  const float* intra  = (const float*)d_in[5];
  const float* inter  = (const float*)d_in[6];
  const float* msg_w1 = (const float*)d_in[7];
  const float* msg_b1 = (const float*)d_in[8];
  const float* msg_w2 = (const float*)d_in[9];
  const float* msg_b2 = (const float*)d_in[10];
  const float* rel_w0 = (const float*)d_in[11];
  const float* rel_w1 = (const float*)d_in[12];
  const float* nw1    = (const float*)d_in[13];
  const float* nb1    = (const float*)d_in[14];
  const float* nw2    = (const float*)d_in[15];
  const float* nb2    = (const float*)d_in[16];
  const float* ew1    = (const float*)d_in[17];
  const float* eb1    = (const float*)d_in[18];
  const float* ew2    = (const float*)d_in[19];
  const float* eb2    = (const float*)d_in[20];
  const float* c0w1   = (const float*)d_in[21];
  const float* c0b1   = (const float*)d_in[22];
  const float* c0w2   = (const float*)d_in[23];
  const float* c1w1   = (const float*)d_in[24];
  const float* c1b1   = (const float*)d_in[25];
  const float* c1w2   = (const float*)d_in[26];
  const float* ln_w   = (const float*)d_in[27];
  const float* ln_b   = (const float*)d_in[28];

  float* ws = (float*)d_ws;
  const size_t OFF_CD   = 0;
  const size_t OFF_RAD  = OFF_CD  + (size_t)NPAIR * 12;
  const size_t OFF_PART = OFF_RAD + (size_t)NPAIR * 16;
  const size_t OFF_RN   = OFF_PART + (size_t)(NPAIR/256) * 16;
  const size_t OFF_M    = OFF_RN  + 16;
  const size_t OFF_AI   = OFF_M   + (size_t)NPAIR * DD;
  const size_t OFF_AE   = OFF_AI  + (size_t)NN * LL * DD;
  const size_t OFF_W    = OFF_AE  + (size_t)NN * LL * DD;
  const size_t OFF_W1T  = OFF_W   + (size_t)NPAIR * CCH;
  const size_t OFF_W2T  = OFF_W1T + (size_t)KMSG * DD;
  const size_t OFF_EW1T = OFF_W2T + (size_t)DD * DD;
  const size_t OFF_EW2T = OFF_EW1T + (size_t)KEDGE * DD;
  const size_t OFF_C0T  = OFF_EW2T + (size_t)DD * PP;
  const size_t OFF_C1T  = OFF_C0T + (size_t)DD * DD;

  float* cd     = ws + OFF_CD;
  float* radial = ws + OFF_RAD;
  float* part   = ws + OFF_PART;
  float* rnorm  = ws + OFF_RN;
  float* m      = ws + OFF_M;
  float* aggi   = ws + OFF_AI;
  float* agge   = ws + OFF_AE;
  float* wgt    = ws + OFF_W;
  float* w1t    = ws + OFF_W1T;
  float* w2t    = ws + OFF_W2T;
  float* ew1t   = ws + OFF_EW1T;
  float* ew2t   = ws + OFF_EW2T;
  float* c0w1t  = ws + OFF_C0T;
  float* c1w1t  = ws + OFF_C1T;

  float* coord_new = (float*)d_out;                    // [N,L,C,3] = 12288
  float* h_new     = coord_new + (size_t)NN*LL*CCH*3;  // [N,L,D]   = 131072
  float* z_new     = h_new + (size_t)NN*LL*DD;         // [N,L,L,P] = 16777216

  // weight transposes ([K][C] -> [C][K]) so B fragments are contiguous b64 loads
  k_transpose<<<(KMSG*DD + 255)/256, 256, 0, stream>>>(msg_w1, w1t, KMSG, DD);
  k_transpose<<<(DD*DD + 255)/256, 256, 0, stream>>>(msg_w2, w2t, DD, DD);
  k_transpose<<<(KEDGE*DD + 255)/256, 256, 0, stream>>>(ew1, ew1t, KEDGE, DD);
  k_transpose<<<(DD*PP + 255)/256, 256, 0, stream>>>(ew2, ew2t, DD, PP);
  k_transpose<<<(DD*DD + 255)/256, 256, 0, stream>>>(c0w1, c0w1t, DD, DD);
  k_transpose<<<(DD*DD + 255)/256, 256, 0, stream>>>(c1w1, c1w1t, DD, DD);

  k_cd_radial<<<NPAIR/256, 256, 0, stream>>>(coord, cd, radial, part);
  k_norm<<<1, 32, 0, stream>>>(part, rnorm);
  k_msg<<<NPAIR/16, 256, 0, stream>>>(h, z, radial, rnorm, w1t, msg_b1, w2t, msg_b2, m);
  k_agg<<<NN*LL, 128, 0, stream>>>(m, intra, inter, aggi, agge);
  k_node<<<NN*LL, 128, 0, stream>>>(h, aggi, agge, rel_w0, rel_w1, nw1, nb1, nw2, nb2, ln_w, ln_b, h_new);
  k_edge<<<NPAIR/16, 256, 0, stream>>>(h_new, z, ew1t, eb1, ew2t, eb2, z_new);
  k_coordw<<<NPAIR/16, 256, 0, stream>>>(m, c0w1t, c0b1, c0w2, c1w1t, c1b1, c1w2, intra, inter, wgt);
  k_coordout<<<NN*LL, 256, 0, stream>>>(coord, cd, wgt, intra, inter, coord_new);

  (void)in_sizes; (void)n_in; (void)out_size; (void)ws_size;
}